// Classifer_86792699117648
// MI455X (gfx1250) — compile-verified
//
#include <hip/hip_runtime.h>
#include <hip/hip_bf16.h>
#include <cstdint>

// ---------------------------------------------------------------------------
// GIN forward for MI455X (gfx1250, wave32).
//  - edge scatter-add: L2-resident (agg = 102MB < 192MB L2) fp32 atomics,
//    compile-time chunk math (no 64-bit divides in the hot 64M-thread launch)
//  - node GEMMs: V_WMMA_F32_16X16X4_F32 (fp32 matrix path = reference
//    precision); W tile staged in LDS PRE-SWIZZLED into B-fragment order so
//    each fragment is one conflict-free ds_load_b64; single WMMA call site
//    with unconditional accumulator dataflow (no loop-phi register shuffles)
//  - BN: two-pass (atomics into per-feature sums), fused normalize+ReLU
//  - head: tiny, naive coalesced kernels; fc2 computes only last 204 cols
// ---------------------------------------------------------------------------

typedef __attribute__((ext_vector_type(2))) float v2f;
typedef __attribute__((ext_vector_type(8))) float v8f;

#define NNODES 100000
#define HID    256
#define NGRAPH 512

// --------------------------- utility: fill ---------------------------------
__global__ void fill_f32(float* __restrict__ p, long long n, float v) {
  long long i = (long long)blockIdx.x * blockDim.x + threadIdx.x;
  long long stride = (long long)gridDim.x * blockDim.x;
  for (; i < n; i += stride) p[i] = v;
}

// --------------------------- edge scatter-add ------------------------------
// one thread per (edge, VEC-feature chunk); CHUNKS is compile-time so the
// e = tid/CHUNKS division strength-reduces. agg is L2-resident: all atomics
// stay on-die.
template <int VEC, int CHUNKS>
__global__ void edge_scatter(const float* __restrict__ h, const int* __restrict__ src,
                             const int* __restrict__ dst, float* __restrict__ agg,
                             int n_edges) {
  constexpr int D = VEC * CHUNKS;
  int tid = blockIdx.x * blockDim.x + threadIdx.x;   // < 64M, fits int32
  if (tid >= n_edges * CHUNKS) return;
  int e = tid / CHUNKS;
  int c = (tid - e * CHUNKS) * VEC;
  int s = src[e];
  int t = dst[e];
  const float* hp = h + (size_t)s * D + c;
  float* ap = agg + (size_t)t * D + c;
  float v[VEC];
#pragma unroll
  for (int i = 0; i < VEC; ++i) v[i] = hp[i];
#pragma unroll
  for (int i = 0; i < VEC; ++i) unsafeAtomicAdd(ap + i, v[i]);
}

// --------------------------- WMMA fp32 GEMM --------------------------------
// y[r, 0:256] = sum_k (hscale*h[r,k] + agg[r,k]) * W[k,n] + bscale*bias[n]
// Block = 256 threads = 8 waves; each wave owns one 16-row M tile, two
// passes of 128 output columns (8 x v8f accumulators = 64 VGPRs).
//
// LDS holds the 32x256 W K-tile pre-swizzled into WMMA B-fragment order:
// fragment row qr = 2*quad + half stores float2 {W[kb,col], W[kb+1,col]}
// with kb = k0 + 4*quad + 2*half. A lane's B fragment is then ONE
// ds_load_b64. Row stride 272 float2 => half-row delta 544 words = 32 banks
// (mod 64); with 2*l16+{0,1} inside a half-wave, one wave's b64 load hits
// all 64 banks exactly once (conflict-free).
//
// The compute path is UNCONDITIONAL: waves past the last M tile use row 0
// (valid memory) and simply never store, so acc[] has a single def chain and
// a single WMMA call site -> no accumulator phi copies across the K loop.
__global__ void __launch_bounds__(256)
gin_gemm_wmma(const float* __restrict__ h, const float* __restrict__ agg,
              const float* __restrict__ W, const float* __restrict__ bias,
              float* __restrict__ y, int nrows, int d_in,
              float hscale, float bscale) {
  constexpr int KT = 32;        // K tile
  constexpr int NQR = KT / 2;   // 16 fragment rows
  constexpr int STR = 272;      // float2 stride per fragment row
  __shared__ v2f wlds[NQR * STR];

  const int lane = threadIdx.x & 31;
  const int wave = threadIdx.x >> 5;  // 0..7
  const int half = lane >> 4;         // K pair offset within quad
  const int l16 = lane & 15;
  const int ntiles = nrows >> 4;       // 100000/16 = 6250
  const int nbt = (ntiles + 7) >> 3;   // block tiles (8 waves each)
  const int kpad = (d_in + KT - 1) / KT * KT;

  for (int bt = blockIdx.x; bt < nbt; bt += gridDim.x) {
    const int mt = bt * 8 + wave;
    const bool active = mt < ntiles;           // wave-uniform (store gate only)
    const int row = active ? (mt * 16 + l16) : 0;  // inactive: valid dummy row
    const size_t rowoff = (size_t)row * d_in;

    for (int np = 0; np < 2; ++np) {
      v8f acc[8] = {};
      for (int k0 = 0; k0 < kpad; k0 += KT) {
        __syncthreads();  // block-uniform loop bounds: no barrier divergence
        // cooperative swizzled fill (zero-padded past d_in; d_in is even)
        for (int j = threadIdx.x; j < NQR * HID; j += 256) {
          int qr = j >> 8;
          int col = j & 255;
          int kb = k0 + ((qr >> 1) << 2) + ((qr & 1) << 1);
          v2f w2 = {};
          if (kb < d_in) {
            w2.x = W[(size_t)kb * HID + col];
            w2.y = W[(size_t)(kb + 1) * HID + col];
          }
          wlds[qr * STR + col] = w2;
        }
        __syncthreads();
        for (int q = 0; q < KT / 4; ++q) {
          const int kg = k0 + q * 4 + 2 * half;
          v2f a = {};
          if (kg < d_in) {  // uniformly true except layer-1 tail K-tile;
                            // d_in even -> float2 never straddles
            float2 hv = *(const float2*)(h + rowoff + kg);
            float2 gv = *(const float2*)(agg + rowoff + kg);
            a.x = hscale * hv.x + gv.x;
            a.y = hscale * hv.y + gv.y;
          }
          const int qr = q * 2 + half;
#pragma unroll
          for (int n = 0; n < 8; ++n) {
            v2f b = wlds[qr * STR + np * 128 + n * 16 + l16];
            acc[n] = __builtin_amdgcn_wmma_f32_16x16x4_f32(
                false, a, false, b, (short)0, acc[n], false, false);
          }
        }
      }
      if (active) {
        // C/D layout: VGPR r -> M = r (lanes 0-15) / r+8 (lanes 16-31), N = l16
        const int rbase = mt * 16 + half * 8;
#pragma unroll
        for (int n = 0; n < 8; ++n) {
          const int col = np * 128 + n * 16 + l16;
          const float bb = bscale * bias[col];
#pragma unroll
          for (int r = 0; r < 8; ++r) {
            y[(size_t)(rbase + r) * HID + col] = acc[n][r] + bb;
          }
        }
      }
    }
  }
}

// --------------------------- BatchNorm (100K rows) -------------------------
__global__ void bn_stats(const float* __restrict__ y, float* __restrict__ st, int nrows) {
  // st[0:256] = sum, st[256:512] = sumsq ; blockDim = 256 (one thread/feature)
  int f = threadIdx.x;
  float s = 0.f, ss = 0.f;
  for (int r = blockIdx.x; r < nrows; r += gridDim.x) {
    float v = y[(size_t)r * HID + f];
    s += v;
    ss += v * v;
  }
  unsafeAtomicAdd(&st[f], s);
  unsafeAtomicAdd(&st[HID + f], ss);
}

__global__ void bn_apply_relu(const float* __restrict__ y, const float* __restrict__ st,
                              const float* __restrict__ g, const float* __restrict__ b,
                              float* __restrict__ out, int nrows) {
  long long i = (long long)blockIdx.x * blockDim.x + threadIdx.x;
  long long n = (long long)nrows * HID;
  long long stride = (long long)gridDim.x * blockDim.x;
  const float inv = 1.0f / (float)nrows;
  for (; i < n; i += stride) {
    int f = (int)(i & (HID - 1));
    float mean = st[f] * inv;
    float var = st[HID + f] * inv - mean * mean;  // biased variance
    float sc = g[f] * rsqrtf(var + 1e-5f);
    float v = (y[i] - mean) * sc + b[f];
    out[i] = v > 0.f ? v : 0.f;
  }
}

// --------------------------- graph readout ---------------------------------
__global__ void readout_sum(const float* __restrict__ h, const int* __restrict__ gid,
                            float* __restrict__ xg, int nrows) {
  int tid = blockIdx.x * blockDim.x + threadIdx.x;  // < 6.4M
  if (tid >= nrows * (HID / 4)) return;
  int node = tid >> 6;            // / (HID/4)
  int c = (tid & 63) * 4;
  int g = gid[node];
  const float* hp = h + (size_t)node * HID + c;
  float* xp = xg + (size_t)g * HID + c;
#pragma unroll
  for (int i = 0; i < 4; ++i) unsafeAtomicAdd(xp + i, hp[i]);
}

// --------------------------- tiny head kernels -----------------------------
__global__ void dense_naive(const float* __restrict__ x, const float* __restrict__ W,
                            const float* __restrict__ b, float* __restrict__ out,
                            int rows, int K, int N, int Wld) {
  int idx = blockIdx.x * blockDim.x + threadIdx.x;
  if (idx >= rows * N) return;
  int r = idx / N;
  int n = idx % N;
  const float* xp = x + (size_t)r * K;
  const float* wp = W + n;
  float acc = b[n];
  for (int k = 0; k < K; ++k) acc += xp[k] * wp[(size_t)k * Wld];
  out[idx] = acc;
}

__global__ void bn_relu_small(float* __restrict__ x, const float* __restrict__ g,
                              const float* __restrict__ b, int rows, int F) {
  int f = blockIdx.x * blockDim.x + threadIdx.x;
  if (f >= F) return;
  float s = 0.f, ss = 0.f;
  for (int r = 0; r < rows; ++r) {
    float v = x[(size_t)r * F + f];
    s += v;
    ss += v * v;
  }
  float mean = s / rows;
  float var = ss / rows - mean * mean;
  float sc = g[f] * rsqrtf(var + 1e-5f);
  float sh = b[f] - mean * sc;
  for (int r = 0; r < rows; ++r) {
    float v = x[(size_t)r * F + f] * sc + sh;
    x[(size_t)r * F + f] = v > 0.f ? v : 0.f;
  }
}

// only the last Nout columns of fc2 survive the reference's slice
__global__ void dense_sigmoid(const float* __restrict__ x, const float* __restrict__ W,
                              const float* __restrict__ b, float* __restrict__ out,
                              int rows, int K, int Ntot, int Nout) {
  int idx = blockIdx.x * blockDim.x + threadIdx.x;
  if (idx >= rows * Nout) return;
  int r = idx / Nout;
  int j = idx % Nout;
  int col = Ntot - Nout + j;
  const float* xp = x + (size_t)r * K;
  const float* wp = W + col;
  float acc = b[col];
  for (int k = 0; k < K; ++k) acc += xp[k] * wp[(size_t)k * Ntot];
  out[idx] = 1.0f / (1.0f + expf(-acc));
}

// ---------------------------------------------------------------------------
extern "C" void kernel_launch(void* const* d_in, const int* in_sizes, int n_in,
                              void* d_out, int out_size, void* d_ws, size_t ws_size,
                              hipStream_t stream) {
  // input order (dict order, list elements flattened):
  const float* h0    = (const float*)d_in[0];   // [100000, 78]
  const int*   src   = (const int*)d_in[1];     // [1M]
  const int*   dst   = (const int*)d_in[2];     // [1M]
  const int*   gid   = (const int*)d_in[3];     // [100000]
  const float* gW[3] = {(const float*)d_in[4], (const float*)d_in[5], (const float*)d_in[6]};
  const float* gB[3] = {(const float*)d_in[7], (const float*)d_in[8], (const float*)d_in[9]};
  const float* bng[3] = {(const float*)d_in[10], (const float*)d_in[11], (const float*)d_in[12]};
  const float* bnb[3] = {(const float*)d_in[13], (const float*)d_in[14], (const float*)d_in[15]};
  const float* fc1_W = (const float*)d_in[16];  // [256,512]
  const float* fc1_b = (const float*)d_in[17];
  const float* bn1_g = (const float*)d_in[18];
  const float* bn1_b = (const float*)d_in[19];
  const float* mlp_W = (const float*)d_in[20];  // [512,256]
  const float* mlp_b = (const float*)d_in[21];
  const float* mbn_g = (const float*)d_in[22];
  const float* mbn_b = (const float*)d_in[23];
  const float* fc2_W = (const float*)d_in[24];  // [256,408]
  const float* fc2_b = (const float*)d_in[25];
  float* out = (float*)d_out;                   // [512,204]

  const int n_edges = in_sizes[1];
  const size_t NH = (size_t)NNODES * HID;       // 25.6M floats
  float* bufA = (float*)d_ws;        // h after BN (layer output)
  float* bufB = bufA + NH;           // y (pre-BN gemm output)
  float* aggb = bufB + NH;           // agg; later overlaid with stats/head scratch
  float* st = aggb;                  // 512 floats, used strictly after gemm reads agg
  float* xg = aggb + 1024;           // [512,256] readout
  float* x1 = xg + (size_t)NGRAPH * HID;  // [512,512]
  float* x2 = x1 + (size_t)NGRAPH * 512;  // [512,256]

  const int dims[4] = {78, HID, HID, HID};
  const float* hcur = h0;

  for (int layer = 0; layer < 3; ++layer) {
    const int d = dims[layer];
    // 1) zero agg
    fill_f32<<<2048, 256, 0, stream>>>(aggb, (long long)NNODES * d, 0.0f);
    // 2) scatter-add h[src] into agg[dst]  (L2-resident atomics)
    if (d == HID) {
      int total = n_edges * (HID / 4);
      edge_scatter<4, HID / 4><<<(total + 255) / 256, 256, 0, stream>>>(
          hcur, src, dst, aggb, n_edges);
    } else {  // d == 78
      int total = n_edges * 39;
      edge_scatter<2, 39><<<(total + 255) / 256, 256, 0, stream>>>(
          hcur, src, dst, aggb, n_edges);
    }
    // 3) y = (2h + agg) @ W + 2b   (fp32 WMMA)
    const int nbt = ((NNODES / 16) + 7) / 8;
    gin_gemm_wmma<<<nbt, 256, 0, stream>>>(hcur, aggb, gW[layer], gB[layer],
                                           bufB, NNODES, d, 2.0f, 2.0f);
    // 4) BN stats (sums/sumsq), then fused normalize+ReLU into bufA
    fill_f32<<<2, 256, 0, stream>>>(st, 2 * HID, 0.0f);
    bn_stats<<<512, 256, 0, stream>>>(bufB, st, NNODES);
    bn_apply_relu<<<4096, 256, 0, stream>>>(bufB, st, bng[layer], bnb[layer],
                                            bufA, NNODES);
    hcur = bufA;
  }

  // readout: xg[g] = sum over nodes in graph g
  fill_f32<<<512, 256, 0, stream>>>(xg, (long long)NGRAPH * HID, 0.0f);
  {
    int total = NNODES * (HID / 4);
    readout_sum<<<(total + 255) / 256, 256, 0, stream>>>(bufA, gid, xg, NNODES);
  }

  // head: fc1 -> BN+ReLU -> mlp -> BN+ReLU -> fc2 (last 204 cols) -> sigmoid
  dense_naive<<<(NGRAPH * 512 + 255) / 256, 256, 0, stream>>>(
      xg, fc1_W, fc1_b, x1, NGRAPH, HID, 512, 512);
  bn_relu_small<<<2, 256, 0, stream>>>(x1, bn1_g, bn1_b, NGRAPH, 512);
  dense_naive<<<(NGRAPH * 256 + 255) / 256, 256, 0, stream>>>(
      x1, mlp_W, mlp_b, x2, NGRAPH, 512, 256, 256);
  bn_relu_small<<<1, 256, 0, stream>>>(x2, mbn_g, mbn_b, NGRAPH, 256);
  dense_sigmoid<<<(NGRAPH * 204 + 255) / 256, 256, 0, stream>>>(
      x2, fc2_W, fc2_b, out, NGRAPH, 256, 408, 204);

  (void)in_sizes; (void)n_in; (void)out_size; (void)ws_size;
}